// OgreModule_79680233276196
// MI455X (gfx1250) — compile-verified
//
#include <hip/hip_runtime.h>
#include <hip/hip_bf16.h>
#include <climits>

typedef __attribute__((ext_vector_type(2))) float v2f;
typedef __attribute__((ext_vector_type(8))) float v8f;

#define N_NODES 20000
#define N_GENESC 2000
#define N_EDGES 640000
#define EMBD 128
#define EDD 32
#define PDD 64
#define OUTD 16

// LDS weight tiles are stored as interleaved K-pairs: element [p][n] is the
// float2 {W[2p][n], W[2p+1][n]} -> one ds_load_b64 yields a finished WMMA B
// fragment. Row stride (in float2 units) is ncols+8 so the lane-group offset
// is 16 DWORDs (8 DWORD-pairs) mod 64: the 32-DWORD spans of lanes 0-15 and 16-31 overlap by 16.
#define KC_EMB 80
#define S2_EMB 136  // 40 pairs * 136 * 8B = 43520 B
#define S2_MSG 72   // 80 pairs * 72  * 8B = 46080 B
#define KC_UPD 96
#define S2_UPD 136  // 48 pairs * 136 * 8B = 52224 B

// D = A(16x4,f32) * B(4x16,f32) + C(16x16,f32)  -> v_wmma_f32_16x16x4_f32
static __device__ __forceinline__ v8f wmma4(v2f a, v2f b, v8f c) {
  return __builtin_amdgcn_wmma_f32_16x16x4_f32(false, a, false, b, (short)0, c,
                                               false, false);
}

static __device__ __forceinline__ void stage_pair(float* dst, const float* r0,
                                                  const float* r1) {
  float4 a = *(const float4*)r0;
  float4 b = *(const float4*)r1;
  float4 lo = {a.x, b.x, a.y, b.y};
  float4 hi = {a.z, b.z, a.w, b.w};
  *(float4*)dst = lo;
  *(float4*)(dst + 4) = hi;
}

// ---------------------------------------------------------------- max(src)
__global__ void init_max_kernel(int* m) {
  if (threadIdx.x == 0 && blockIdx.x == 0) *m = INT_MIN;
}

__global__ void max_src_kernel(const int* __restrict__ src, int n, int* m) {
  int v = INT_MIN;
  for (int i = blockIdx.x * blockDim.x + threadIdx.x; i < n;
       i += gridDim.x * blockDim.x)
    v = max(v, src[i]);
  for (int off = 16; off > 0; off >>= 1) v = max(v, __shfl_down(v, off, 32));
  if ((threadIdx.x & 31) == 0) atomicMax(m, v);
}

__global__ void zero_kernel(float* p, int n) {
  for (int i = blockIdx.x * blockDim.x + threadIdx.x; i < n;
       i += gridDim.x * blockDim.x)
    p[i] = 0.0f;
}

// ------------------------------------------------- h = x @ W_emb + b_emb
// Wave computes 16 rows x 128 cols (8 accums); block (8 waves) = 128 rows.
__global__ void __launch_bounds__(256, 1)
emb_gemm_kernel(const float* __restrict__ x, const float* __restrict__ W,
                const float* __restrict__ b, float* __restrict__ h) {
  __shared__ float2 sW[(KC_EMB / 2) * S2_EMB];
  const int lane = threadIdx.x & 31;
  const int wv = threadIdx.x >> 5;
  const int l15 = lane & 15;
  const int half = lane >> 4;                 // 0: K0/K1, 1: K2/K3
  const int m0 = blockIdx.x * 128 + wv * 16;  // row tile base

  int arow = m0 + l15;
  if (arow >= N_NODES) arow = N_NODES - 1;    // clamp loads; stores guarded
  const float* __restrict__ ar = x + (long)arow * N_GENESC;

  v8f c[8] = {};
  for (int k0 = 0; k0 < N_GENESC; k0 += KC_EMB) {
    __syncthreads();  // previous chunk fully consumed
#pragma unroll
    for (int t = 0; t < 5; ++t) {  // 40 pairs * 32 colgroups / 256 threads
      const int g = threadIdx.x + t * 256;
      const int p = g >> 5;
      const int nc = (g & 31) * 4;
      const float* r0 = &W[(long)(k0 + 2 * p) * EMBD + nc];
      stage_pair((float*)&sW[p * S2_EMB + nc], r0, r0 + EMBD);
    }
    __syncthreads();
    __builtin_prefetch(ar + k0 + KC_EMB, 0, 1);
    for (int k = 0; k < KC_EMB; k += 4) {
      v2f a = *(const v2f*)(ar + k0 + k + 2 * half);
      const int p = (k >> 1) + half;
#pragma unroll
      for (int j = 0; j < 8; ++j) {
        float2 t = sW[p * S2_EMB + j * 16 + l15];
        v2f bf = {t.x, t.y};
        c[j] = wmma4(a, bf, c[j]);
      }
    }
  }
#pragma unroll
  for (int r = 0; r < 8; ++r) {
    const int row = m0 + r + 8 * half;
    if (row < N_NODES) {
#pragma unroll
      for (int j = 0; j < 8; ++j) {
        const int n = j * 16 + l15;
        h[(long)row * EMBD + n] = c[j][r] + b[n];
      }
    }
  }
}

// --------------- m = relu([h[src],ea] @ mW + mb) * P[idx]; scatter-add to aggr
// Wave: 16 edges x 64 (all PD) cols; block = 128 edges. mW staged in LDS.
__global__ void __launch_bounds__(256, 2)
msg_gemm_kernel(const float* __restrict__ h, const float* __restrict__ ea,
                const int* __restrict__ src, const int* __restrict__ dst,
                const float* __restrict__ mW, const float* __restrict__ mb,
                const float* __restrict__ P, const int* __restrict__ maxbuf,
                float* __restrict__ aggr) {
  __shared__ float2 sW[((EMBD + EDD) / 2) * S2_MSG];
  const int lane = threadIdx.x & 31;
  const int wv = threadIdx.x >> 5;
  const int l15 = lane & 15;
  const int half = lane >> 4;
  const long e0 = (long)blockIdx.x * 128 + wv * 16;

#pragma unroll
  for (int t = 0; t < 5; ++t) {  // 80 pairs * 16 colgroups / 256 threads
    const int g = threadIdx.x + t * 256;
    const int p = g >> 4;
    const int nc = (g & 15) * 4;
    const float* r0 = &mW[(long)(2 * p) * PDD + nc];
    stage_pair((float*)&sW[p * S2_MSG + nc], r0, r0 + PDD);
  }

  long er = e0 + l15;
  if (er >= N_EDGES) er = N_EDGES - 1;
  const int s = src[er];
  const float* __restrict__ hr = h + (long)s * EMBD;
  const float* __restrict__ eR = ea + er * EDD;
  __syncthreads();

  v8f c[4] = {};
  // K = 0..127 comes from gathered h[src]
  for (int k = 0; k < EMBD; k += 4) {
    v2f a = *(const v2f*)(hr + k + 2 * half);
    const int p = (k >> 1) + half;
#pragma unroll
    for (int j = 0; j < 4; ++j) {
      float2 t = sW[p * S2_MSG + j * 16 + l15];
      v2f bf = {t.x, t.y};
      c[j] = wmma4(a, bf, c[j]);
    }
  }
  // K = 128..159 comes from edge_attr
  for (int k = EMBD; k < EMBD + EDD; k += 4) {
    v2f a = *(const v2f*)(eR + (k - EMBD) + 2 * half);
    const int p = (k >> 1) + half;
#pragma unroll
    for (int j = 0; j < 4; ++j) {
      float2 t = sW[p * S2_MSG + j * 16 + l15];
      v2f bf = {t.x, t.y};
      c[j] = wmma4(a, bf, c[j]);
    }
  }

  const int off = *maxbuf - (N_GENESC - 1);
#pragma unroll
  for (int r = 0; r < 8; ++r) {
    const long erow = e0 + r + 8 * half;
    if (erow < N_EDGES) {
      const int d = dst[erow];
      const int g = src[erow] - off;
#pragma unroll
      for (int j = 0; j < 4; ++j) {
        const int n = j * 16 + l15;
        float v = c[j][r] + mb[n];
        v = v > 0.0f ? v : 0.0f;
        v *= P[(long)g * PDD + n];
        atomicAdd(&aggr[(long)d * PDD + n], v);
      }
    }
  }
}

// ---------------------------- h' = relu([aggr,h] @ uW + ub)   (K = 192)
// Wave: 16 rows x 128 cols (8 accums). uW staged in LDS in 2 chunks of 96.
__global__ void __launch_bounds__(256, 1)
upd_gemm_kernel(const float* __restrict__ aggr, const float* __restrict__ h,
                const float* __restrict__ uW, const float* __restrict__ ub,
                float* __restrict__ hout) {
  __shared__ float2 sW[(KC_UPD / 2) * S2_UPD];
  const int lane = threadIdx.x & 31;
  const int wv = threadIdx.x >> 5;
  const int l15 = lane & 15;
  const int half = lane >> 4;
  const int m0 = blockIdx.x * 128 + wv * 16;

  int arow = m0 + l15;
  if (arow >= N_NODES) arow = N_NODES - 1;
  const float* __restrict__ gr = aggr + (long)arow * PDD;
  const float* __restrict__ hr = h + (long)arow * EMBD;

  v8f c[8] = {};
  for (int k0 = 0; k0 < PDD + EMBD; k0 += KC_UPD) {
    __syncthreads();
#pragma unroll
    for (int t = 0; t < 6; ++t) {  // 48 pairs * 32 colgroups / 256 threads
      const int g = threadIdx.x + t * 256;
      const int p = g >> 5;
      const int nc = (g & 31) * 4;
      const float* r0 = &uW[(long)(k0 + 2 * p) * EMBD + nc];
      stage_pair((float*)&sW[p * S2_UPD + nc], r0, r0 + EMBD);
    }
    __syncthreads();
    for (int k = 0; k < KC_UPD; k += 4) {
      const int kg = k0 + k;  // wave-uniform; source switch at kg == 64
      v2f a = (kg < PDD) ? *(const v2f*)(gr + kg + 2 * half)
                         : *(const v2f*)(hr + (kg - PDD) + 2 * half);
      const int p = (k >> 1) + half;
#pragma unroll
      for (int j = 0; j < 8; ++j) {
        float2 t = sW[p * S2_UPD + j * 16 + l15];
        v2f bf = {t.x, t.y};
        c[j] = wmma4(a, bf, c[j]);
      }
    }
  }
#pragma unroll
  for (int r = 0; r < 8; ++r) {
    const int row = m0 + r + 8 * half;
    if (row < N_NODES) {
#pragma unroll
      for (int j = 0; j < 8; ++j) {
        const int n = j * 16 + l15;
        float v = c[j][r] + ub[n];
        hout[(long)row * EMBD + n] = v > 0.0f ? v : 0.0f;
      }
    }
  }
}

// ---------------------------------------- node_pred = h @ W_node + b_node
__global__ void node_pred_kernel(const float* __restrict__ h,
                                 const float* __restrict__ Wn,
                                 const float* __restrict__ bn,
                                 float* __restrict__ out) {
  const int lane = threadIdx.x & 31;
  const int wv = threadIdx.x >> 5;
  const int l15 = lane & 15;
  const int half = lane >> 4;
  const int m0 = blockIdx.x * 128 + wv * 16;

  int arow = m0 + l15;
  if (arow >= N_NODES) arow = N_NODES - 1;
  const float* __restrict__ hr = h + (long)arow * EMBD;

  v8f c = {};
  for (int k = 0; k < EMBD; k += 4) {
    v2f a = *(const v2f*)(hr + k + 2 * half);
    const int kk = k + 2 * half;
    const int n = l15;
    v2f bf = {Wn[kk * OUTD + n], Wn[(kk + 1) * OUTD + n]};
    c = wmma4(a, bf, c);
  }
#pragma unroll
  for (int r = 0; r < 8; ++r) {
    const int row = m0 + r + 8 * half;
    if (row < N_NODES) out[(long)row * OUTD + l15] = c[r] + bn[l15];
  }
}

// ------------------- edge_pred = [h[src],h[dst],ea] . W_ep + b_ep  (mat-vec)
__global__ void edge_pred_kernel(const float* __restrict__ h,
                                 const int* __restrict__ src,
                                 const int* __restrict__ dst,
                                 const float* __restrict__ ea,
                                 const float* __restrict__ Wep,
                                 const float* __restrict__ bep,
                                 float* __restrict__ out) {
  const int e = blockIdx.x * blockDim.x + threadIdx.x;
  if (e >= N_EDGES) return;
  const int s = src[e], d = dst[e];
  const float4* hs = (const float4*)(h + (long)s * EMBD);
  const float4* hd = (const float4*)(h + (long)d * EMBD);
  const float4* er = (const float4*)(ea + (long)e * EDD);
  const float4* w0 = (const float4*)(Wep);
  const float4* w1 = (const float4*)(Wep + EMBD);
  const float4* w2 = (const float4*)(Wep + 2 * EMBD);
  float acc = bep[0];
#pragma unroll 8
  for (int i = 0; i < EMBD / 4; ++i) {
    float4 a = hs[i], w = w0[i];
    acc += a.x * w.x + a.y * w.y + a.z * w.z + a.w * w.w;
  }
#pragma unroll 8
  for (int i = 0; i < EMBD / 4; ++i) {
    float4 a = hd[i], w = w1[i];
    acc += a.x * w.x + a.y * w.y + a.z * w.z + a.w * w.w;
  }
#pragma unroll
  for (int i = 0; i < EDD / 4; ++i) {
    float4 a = er[i], w = w2[i];
    acc += a.x * w.x + a.y * w.y + a.z * w.z + a.w * w.w;
  }
  out[e] = acc;
}

extern "C" void kernel_launch(void* const* d_in, const int* in_sizes, int n_in,
                              void* d_out, int out_size, void* d_ws,
                              size_t ws_size, hipStream_t stream) {
  const float* x = (const float*)d_in[0];       // [20000,2000]
  const float* ea = (const float*)d_in[1];      // [640000,32]
  const int* ei = (const int*)d_in[2];          // [2,640000]
  const float* W_emb = (const float*)d_in[3];   // [2000,128]
  const float* b_emb = (const float*)d_in[4];   // [128]
  const float* msg_W = (const float*)d_in[5];   // [2,160,64]
  const float* msg_b = (const float*)d_in[6];   // [2,64]
  const float* upd_W = (const float*)d_in[7];   // [2,192,128]
  const float* upd_b = (const float*)d_in[8];   // [2,128]
  const float* W_node = (const float*)d_in[9];  // [128,16]
  const float* b_node = (const float*)d_in[10]; // [16]
  const float* W_ep = (const float*)d_in[11];   // [288,1]
  const float* b_ep = (const float*)d_in[12];   // [1]
  const float* P = (const float*)d_in[13];      // [2000,64]

  const int* srcp = ei;
  const int* dstp = ei + N_EDGES;

  float* ws = (float*)d_ws;
  float* hA = ws;                                    // 20000*128
  float* hB = hA + (size_t)N_NODES * EMBD;           // 20000*128
  float* aggr = hB + (size_t)N_NODES * EMBD;         // 20000*64
  int* maxb = (int*)(aggr + (size_t)N_NODES * PDD);  // 1 int

  // max(src) for the gene-index offset
  init_max_kernel<<<1, 1, 0, stream>>>(maxb);
  max_src_kernel<<<256, 256, 0, stream>>>(srcp, N_EDGES, maxb);

  // embedding GEMM
  const int mblocks = (N_NODES + 127) / 128;  // 157
  emb_gemm_kernel<<<mblocks, 256, 0, stream>>>(x, W_emb, b_emb, hA);

  float* hcur = hA;
  float* hnext = hB;
  for (int l = 0; l < 2; ++l) {
    zero_kernel<<<1024, 256, 0, stream>>>(aggr, N_NODES * PDD);
    msg_gemm_kernel<<<N_EDGES / 128, 256, 0, stream>>>(
        hcur, ea, srcp, dstp, msg_W + (size_t)l * (EMBD + EDD) * PDD,
        msg_b + (size_t)l * PDD, P, maxb, aggr);
    upd_gemm_kernel<<<mblocks, 256, 0, stream>>>(
        aggr, hcur, upd_W + (size_t)l * (PDD + EMBD) * EMBD,
        upd_b + (size_t)l * EMBD, hnext);
    float* t = hcur; hcur = hnext; hnext = t;
  }

  float* node_out = (float*)d_out;                 // [20000,16]
  float* edge_out = node_out + N_NODES * OUTD;     // [640000,1]
  node_pred_kernel<<<mblocks, 256, 0, stream>>>(hcur, W_node, b_node, node_out);
  edge_pred_kernel<<<(N_EDGES + 255) / 256, 256, 0, stream>>>(
      hcur, srcp, dstp, ea, W_ep, b_ep, edge_out);
}